// RSSM_78469052498394
// MI455X (gfx1250) — compile-verified
//
#include <hip/hip_runtime.h>
#include <math.h>

// ---------------------------------------------------------------------------
// RSSM scan (DreamerV3-style) for MI455X / gfx1250.
// f32 WMMA (V_WMMA_F32_16X16X4_F32) GEMMs, templated on N so all B-matrix
// loads use immediate offsets from one base pointer (no spills, minimal
// address VALU). Weights (~21MB) stay L2-resident across the 64 sequential
// steps -> matrix-pipe bound. Edge strips are shifted (n0 = min(n0, N-64))
// so every load/store is in-bounds and guard-free; overlapping columns are
// recomputed with identical values.
// ---------------------------------------------------------------------------

typedef __attribute__((ext_vector_type(2))) float v2f;
typedef __attribute__((ext_vector_type(8))) float v8f;

static constexpr int B    = 256;
static constexpr int T    = 64;
static constexpr int ACT  = 6;
static constexpr int EMB  = 1024;
static constexpr int S    = 32;
static constexpr int D    = 32;
static constexpr int SF   = S * D;         // 1024
static constexpr int DET  = 600;
static constexpr int HID  = 600;
static constexpr int OUTW = 4 * SF + DET;  // 4696
static constexpr int HIDP = 608;           // HID padded to 16
static constexpr int DETP = 608;           // DET padded to 16
static constexpr int G3P  = 1808;          // 3*DET=1800 padded to 16

#define WMMA_ACC(AX, BX, CC)                                                   \
  CC = __builtin_amdgcn_wmma_f32_16x16x4_f32(false, (AX), false, (BX),         \
                                             (short)0, (CC), false, false)

// ---------------------------------------------------------------------------
// out[M=256, N] = act( [A1 (K1) | A2 (K2)] @ W[(K1+K2) x N] + bias )
// Compile-time N/LDOUT/ACTMODE. Each wave: 32x64 output strip (2 M-tiles x
// 4 N-tiles, 8 f32 accumulators). K1 % 4 == 0 at all call sites.
// f32 16x16x4 fragment layouts (ISA 7.12.2):
//   A: lane&15 = M row; VGPR {0,1} = K {koff, koff+1}, koff = (lane>>4)*2
//   B: lane&15 = N col; VGPR {0,1} = K {koff, koff+1}   (symmetric)
//   C/D: VGPR v -> M = v + 8*(lane>>4), N = lane&15
// ---------------------------------------------------------------------------
template <int N, int LDOUT, int ACTMODE>
__global__ void gemm_wmma(const float* __restrict__ A1, int sA1, int K1,
                          const float* __restrict__ A2, int sA2, int K2,
                          const float* __restrict__ W,
                          const float* __restrict__ bias,
                          float* __restrict__ out, int totalStrips) {
  int strip = blockIdx.x * 8 + (threadIdx.x >> 5);  // 8 waves / 256-thread blk
  if (strip >= totalStrips) return;
  int lane  = threadIdx.x & 31;
  int mp    = strip & 7;                      // M-pair id: tiles 2mp, 2mp+1
  int n0    = min((strip >> 3) * 64, N - 64); // shifted edge strip (in-bounds)
  int col   = lane & 15;
  int half  = lane >> 4;
  int koff  = half * 2;                       // 0 or 2
  int m0row = (2 * mp) * 16 + col;
  int m1row = (2 * mp + 1) * 16 + col;

  const float* ap0 = A1 + (long)m0row * sA1 + koff;
  const float* ap1 = A1 + (long)m1row * sA1 + koff;
  const float* bp  = W + (long)koff * N + n0 + col;  // single B base pointer

  v8f c00 = {}, c01 = {}, c02 = {}, c03 = {};
  v8f c10 = {}, c11 = {}, c12 = {}, c13 = {};

  // ---- main region: A1, fully unguarded ----
  for (int k = 0; k < K1; k += 4) {
    v2f a0 = *(const v2f*)ap0; ap0 += 4;
    v2f a1 = *(const v2f*)ap1; ap1 += 4;
    v2f b0; b0.x = bp[0];  b0.y = bp[N];        // immediate offsets (N const)
    v2f b1; b1.x = bp[16]; b1.y = bp[N + 16];
    v2f b2; b2.x = bp[32]; b2.y = bp[N + 32];
    v2f b3; b3.x = bp[48]; b3.y = bp[N + 48];
    bp += 4 * N;
    WMMA_ACC(a0, b0, c00); WMMA_ACC(a1, b0, c10);
    WMMA_ACC(a0, b1, c01); WMMA_ACC(a1, b1, c11);
    WMMA_ACC(a0, b2, c02); WMMA_ACC(a1, b2, c12);
    WMMA_ACC(a0, b3, c03); WMMA_ACC(a1, b3, c13);
  }

  // ---- second region: A2 ----
  if (K2 > 0) {
    const float* a2p0 = A2 + (long)m0row * sA2 + koff;
    const float* a2p1 = A2 + (long)m1row * sA2 + koff;
    if ((K2 & 3) == 0) {
      // aligned (embed, K2=1024): unguarded, B pointer continues past K1 rows
      for (int k = 0; k < K2; k += 4) {
        v2f a0 = *(const v2f*)a2p0; a2p0 += 4;
        v2f a1 = *(const v2f*)a2p1; a2p1 += 4;
        v2f b0; b0.x = bp[0];  b0.y = bp[N];
        v2f b1; b1.x = bp[16]; b1.y = bp[N + 16];
        v2f b2; b2.x = bp[32]; b2.y = bp[N + 32];
        v2f b3; b3.x = bp[48]; b3.y = bp[N + 48];
        bp += 4 * N;
        WMMA_ACC(a0, b0, c00); WMMA_ACC(a1, b0, c10);
        WMMA_ACC(a0, b1, c01); WMMA_ACC(a1, b1, c11);
        WMMA_ACC(a0, b2, c02); WMMA_ACC(a1, b2, c12);
        WMMA_ACC(a0, b3, c03); WMMA_ACC(a1, b3, c13);
      }
    } else {
      // tiny guarded tail (action, K2=6): at most 2 iterations
      for (int k2 = 0; k2 < K2; k2 += 4) {
        int kb = k2 + koff;
        bool ok0 = kb < K2, ok1 = (kb + 1) < K2;
        v2f a0, a1;
        a0.x = ok0 ? a2p0[0] : 0.0f; a0.y = ok1 ? a2p0[1] : 0.0f; a2p0 += 4;
        a1.x = ok0 ? a2p1[0] : 0.0f; a1.y = ok1 ? a2p1[1] : 0.0f; a2p1 += 4;
        v2f b0; b0.x = ok0 ? bp[0]  : 0.0f; b0.y = ok1 ? bp[N]      : 0.0f;
        v2f b1; b1.x = ok0 ? bp[16] : 0.0f; b1.y = ok1 ? bp[N + 16] : 0.0f;
        v2f b2; b2.x = ok0 ? bp[32] : 0.0f; b2.y = ok1 ? bp[N + 32] : 0.0f;
        v2f b3; b3.x = ok0 ? bp[48] : 0.0f; b3.y = ok1 ? bp[N + 48] : 0.0f;
        bp += 4 * N;
        WMMA_ACC(a0, b0, c00); WMMA_ACC(a1, b0, c10);
        WMMA_ACC(a0, b1, c01); WMMA_ACC(a1, b1, c11);
        WMMA_ACC(a0, b2, c02); WMMA_ACC(a1, b2, c12);
        WMMA_ACC(a0, b3, c03); WMMA_ACC(a1, b3, c13);
      }
    }
  }

  // ---- epilogue: bias + activation + store (all in-bounds, no guards) ----
  v8f accs[8] = {c00, c01, c02, c03, c10, c11, c12, c13};
#pragma unroll
  for (int q = 0; q < 8; ++q) {
    int mt = (q < 4) ? (2 * mp) : (2 * mp + 1);
    int n  = n0 + (q & 3) * 16 + col;
    float bv = bias[n];
    float* op = out + (long)(mt * 16 + 8 * half) * LDOUT + n;
#pragma unroll
    for (int v = 0; v < 8; ++v) {
      float val = accs[q][v] + bv;
      if (ACTMODE == 1) val = (val > 0.0f) ? val : expm1f(val);  // ELU
      op[(long)v * LDOUT] = val;
    }
  }
}

// ---------------------------------------------------------------------------
// GRU (keras reset_after) gate fusion: deter_n = z*deter + (1-z)*cand.
// Also writes deter_n into the output tensor's deter section.
// ---------------------------------------------------------------------------
__global__ void gru_gate(const float* __restrict__ gx, const float* __restrict__ gh,
                         const float* __restrict__ deter_in,
                         float* __restrict__ deter_out,
                         float* __restrict__ dout, int t) {
  int idx = blockIdx.x * blockDim.x + threadIdx.x;
  if (idx >= B * DETP) return;
  int m = idx / DETP;
  int j = idx % DETP;
  if (j >= DET) { deter_out[(long)m * DETP + j] = 0.0f; return; }
  float xz = gx[(long)m * G3P + j];
  float xr = gx[(long)m * G3P + DET + j];
  float xh = gx[(long)m * G3P + 2 * DET + j];
  float hz = gh[(long)m * G3P + j];
  float hr = gh[(long)m * G3P + DET + j];
  float hh = gh[(long)m * G3P + 2 * DET + j];
  float z    = 1.0f / (1.0f + expf(-(xz + hz)));
  float r    = 1.0f / (1.0f + expf(-(xr + hr)));
  float cand = tanhf(xh + r * hh);
  float dprev = deter_in[(long)m * DETP + j];
  float dn = z * dprev + (1.0f - z) * cand;
  deter_out[(long)m * DETP + j] = dn;
  dout[((long)m * T + t) * OUTW + 2 * SF + j] = dn;
}

// ---------------------------------------------------------------------------
// Gumbel-max categorical sample w/ straight-through forward value.
// One thread per (batch, stoch-group): argmax(logit+g) over D=32, softmax
// with max-subtraction (matches jax.nn.softmax).
// ---------------------------------------------------------------------------
__global__ void sample_write(const float* __restrict__ logit,   // [B x SF]
                             const float* __restrict__ noise,   // [B,T,S,D]
                             int t, float* __restrict__ dout,
                             int col_logit, int col_stoch,
                             float* __restrict__ stoch_carry) { // nullable
  int id = blockIdx.x * blockDim.x + threadIdx.x;
  if (id >= B * S) return;
  int m = id / S;
  int s = id % S;
  const float* lrow = logit + (long)m * SF + s * D;
  const float* grow = noise + (((long)m * T + t) * S + s) * D;

  float best = -INFINITY, mx = -INFINITY;
  int bi = 0;
  for (int d = 0; d < D; ++d) {
    float lv = lrow[d];
    float pg = lv + grow[d];
    if (pg > best) { best = pg; bi = d; }  // first max wins (matches argmax)
    if (lv > mx) mx = lv;
  }
  float sum = 0.0f;
  for (int d = 0; d < D; ++d) sum += expf(lrow[d] - mx);
  float inv = 1.0f / sum;

  long orow = ((long)m * T + t) * OUTW;
  for (int d = 0; d < D; ++d) {
    float lv = lrow[d];
    float p  = expf(lv - mx) * inv;
    float oh = (d == bi) ? 1.0f : 0.0f;
    float st = (oh + p) - p;               // same FP ordering as reference
    dout[orow + col_logit + s * D + d] = lv;
    dout[orow + col_stoch + s * D + d] = st;
    if (stoch_carry) stoch_carry[(long)m * SF + s * D + d] = st;
  }
}

// ---------------------------------------------------------------------------
// Host-side launch
// ---------------------------------------------------------------------------
static inline int gemmBlocks(int N, int& totalStrips) {
  int stripsN = (N + 63) / 64;         // 64 output cols per wave (shifted edge)
  totalStrips = 8 * stripsN;           // 8 M-pairs (32 rows each)
  return (totalStrips + 7) / 8;        // 8 waves per 256-thread block
}

extern "C" void kernel_launch(void* const* d_in, const int* in_sizes, int n_in,
                              void* d_out, int out_size, void* d_ws, size_t ws_size,
                              hipStream_t stream) {
  (void)in_sizes; (void)n_in; (void)out_size; (void)ws_size;
  const float* action = (const float*)d_in[0];
  const float* embed  = (const float*)d_in[1];
  const float* g_img  = (const float*)d_in[2];
  const float* g_obs  = (const float*)d_in[3];
  const float* W_img1 = (const float*)d_in[4];
  const float* b_img1 = (const float*)d_in[5];
  const float* Wx     = (const float*)d_in[6];
  const float* Wh     = (const float*)d_in[7];
  const float* b_in   = (const float*)d_in[8];
  const float* b_rec  = (const float*)d_in[9];
  const float* W_img2 = (const float*)d_in[10];
  const float* b_img2 = (const float*)d_in[11];
  const float* W_img5 = (const float*)d_in[12];
  const float* b_img5 = (const float*)d_in[13];
  const float* W_obs1 = (const float*)d_in[14];
  const float* b_obs1 = (const float*)d_in[15];
  const float* W_obs3 = (const float*)d_in[16];
  const float* b_obs3 = (const float*)d_in[17];
  float* out = (float*)d_out;

  // Workspace carve-up (floats)
  float* w = (float*)d_ws;
  float* stoch  = w;  w += (long)B * SF;     // carry
  float* deterA = w;  w += (long)B * DETP;   // carry ping
  float* deterB = w;  w += (long)B * DETP;   // carry pong
  float* x      = w;  w += (long)B * HIDP;
  float* gx     = w;  w += (long)B * G3P;
  float* gh     = w;  w += (long)B * G3P;
  float* y      = w;  w += (long)B * HIDP;
  float* lp     = w;  w += (long)B * SF;     // prior logits
  float* o      = w;  w += (long)B * HIDP;
  float* lq     = w;  w += (long)B * SF;     // post logits

  // zero initial state (RSSM.initial)
  hipMemsetAsync(stoch,  0, sizeof(float) * (size_t)B * SF,   stream);
  hipMemsetAsync(deterA, 0, sizeof(float) * (size_t)B * DETP, stream);

  int ts600, ts1800, ts1024;
  const int blk600  = gemmBlocks(HID,     ts600);   // N=600
  const int blk1800 = gemmBlocks(3 * DET, ts1800);  // N=1800
  const int blk1024 = gemmBlocks(SF,      ts1024);  // N=1024
  const int gateBlocks   = (B * DETP + 255) / 256;
  const int sampleBlocks = (B * S + 255) / 256;

  for (int t = 0; t < T; ++t) {
    float* d_in_  = (t & 1) ? deterB : deterA;
    float* d_out_ = (t & 1) ? deterA : deterB;

    // x = elu([stoch | action_t] @ W_img1 + b_img1)
    gemm_wmma<HID, HIDP, 1><<<blk600, 256, 0, stream>>>(
        stoch, SF, SF, action + (long)t * ACT, T * ACT, ACT,
        W_img1, b_img1, x, ts600);

    // gx = x @ Wx + b_in ; gh = deter @ Wh + b_rec
    gemm_wmma<3 * DET, G3P, 0><<<blk1800, 256, 0, stream>>>(
        x, HIDP, HID, nullptr, 0, 0, Wx, b_in, gx, ts1800);
    gemm_wmma<3 * DET, G3P, 0><<<blk1800, 256, 0, stream>>>(
        d_in_, DETP, DET, nullptr, 0, 0, Wh, b_rec, gh, ts1800);

    // GRU gates -> deter_n (also stored to output)
    gru_gate<<<gateBlocks, 256, 0, stream>>>(gx, gh, d_in_, d_out_, out, t);

    // y = elu(deter_n @ W_img2 + b_img2) ; logit_prior = y @ W_img5 + b_img5
    gemm_wmma<HID, HIDP, 1><<<blk600, 256, 0, stream>>>(
        d_out_, DETP, DET, nullptr, 0, 0, W_img2, b_img2, y, ts600);
    gemm_wmma<SF, SF, 0><<<blk1024, 256, 0, stream>>>(
        y, HIDP, HID, nullptr, 0, 0, W_img5, b_img5, lp, ts1024);

    // prior sample -> output cols [2648, 3672) logits, [3672, 4696) sample
    sample_write<<<sampleBlocks, 256, 0, stream>>>(
        lp, g_img, t, out, 2 * SF + DET, 3 * SF + DET, nullptr);

    // o = elu([deter_n | embed_t] @ W_obs1 + b_obs1); logit_post = o @ W_obs3 + b_obs3
    gemm_wmma<HID, HIDP, 1><<<blk600, 256, 0, stream>>>(
        d_out_, DETP, DET, embed + (long)t * EMB, T * EMB, EMB,
        W_obs1, b_obs1, o, ts600);
    gemm_wmma<SF, SF, 0><<<blk1024, 256, 0, stream>>>(
        o, HIDP, HID, nullptr, 0, 0, W_obs3, b_obs3, lq, ts1024);

    // post sample -> output cols [0,1024) logits, [1024,2048) sample + carry
    sample_write<<<sampleBlocks, 256, 0, stream>>>(
        lq, g_obs, t, out, 0, SF, stoch);
  }
}